// DGASEncoder_61280593379865
// MI455X (gfx1250) — compile-verified
//
#include <hip/hip_runtime.h>
#include <hip/hip_bf16.h>

// Problem constants (from reference)
#define B_ 4
#define N_ 4096
#define C_ 128
#define K_ 32
#define RADIUS2 0.01f     // 0.1^2
#define BN_EPS 1e-5f

typedef __attribute__((ext_vector_type(2))) float v2f;
typedef __attribute__((ext_vector_type(4))) float v4f;
typedef __attribute__((ext_vector_type(8))) float v8f;

// ---------------------------------------------------------------------------
// Kernel 1: h[b,o,n] = sum_c W1[o,c] * f[b,c,n] + b1[o]
// One wave (32 lanes) computes one 16x16 output tile with V_WMMA_F32_16X16X4_F32.
// A-frag layout (16x4 f32): lane<16: VGPR0=A[row][k0], VGPR1=A[row][k0+1];
//                           lane>=16: K offset +2.   (ISA 7.12.2)
// C/D layout: VGPR r, lane L -> M = r + 8*(L>=16), N = L%16.
// ---------------------------------------------------------------------------
__global__ void __launch_bounds__(32) gemm_wmma(const float* __restrict__ W1,
                                                const float* __restrict__ f,
                                                const float* __restrict__ b1,
                                                float* __restrict__ h) {
  const int n0   = blockIdx.x * 16;
  const int m0   = blockIdx.y * 16;
  const int b    = blockIdx.z;
  const int lane = threadIdx.x;
  const int row  = lane & 15;
  const int hi   = lane >> 4;          // 0 for lanes 0-15, 1 for lanes 16-31

  const float* __restrict__ fb = f + (size_t)b * C_ * N_;
  const float* __restrict__ wrow = W1 + (size_t)(m0 + row) * C_;

  v8f acc = {};
#pragma unroll 4
  for (int k0 = 0; k0 < C_; k0 += 4) {
    const int ka = k0 + 2 * hi;
    v2f a, bfrag;
    a.x = wrow[ka];
    a.y = wrow[ka + 1];
    bfrag.x = fb[(size_t)ka * N_ + n0 + row];
    bfrag.y = fb[(size_t)(ka + 1) * N_ + n0 + row];
    acc = __builtin_amdgcn_wmma_f32_16x16x4_f32(
        /*neg_a=*/false, a, /*neg_b=*/false, bfrag,
        /*c_mod=*/(short)0, acc, /*reuse_a=*/false, /*reuse_b=*/false);
  }

  float* __restrict__ hb = h + (size_t)b * C_ * N_;
#pragma unroll
  for (int r = 0; r < 8; ++r) {
    const int m = m0 + r + 8 * hi;
    hb[(size_t)m * N_ + n0 + row] = acc[r] + b1[m];
  }
}

// ---------------------------------------------------------------------------
// Kernel 2: per-channel mean/var over (B,N); emit fused affine:
//   scale[c] = gamma[c] * rsqrt(var+eps),  shift[c] = beta[c] - mean*scale[c]
// ---------------------------------------------------------------------------
__global__ void __launch_bounds__(256) bn_stats(const float* __restrict__ h,
                                                const float* __restrict__ gamma,
                                                const float* __restrict__ beta,
                                                float* __restrict__ scale,
                                                float* __restrict__ shift) {
  const int c   = blockIdx.x;
  const int tid = threadIdx.x;
  float s = 0.f, s2 = 0.f;
  for (int b = 0; b < B_; ++b) {
    const float* __restrict__ row = h + ((size_t)b * C_ + c) * N_;
    for (int n = tid; n < N_; n += 256) {
      const float v = row[n];
      s += v;
      s2 += v * v;
    }
  }
  __shared__ float ls[256], ls2[256];
  ls[tid] = s; ls2[tid] = s2;
  __syncthreads();
  for (int off = 128; off > 0; off >>= 1) {
    if (tid < off) { ls[tid] += ls[tid + off]; ls2[tid] += ls2[tid + off]; }
    __syncthreads();
  }
  if (tid == 0) {
    const float inv_cnt = 1.0f / (float)(B_ * N_);
    const float mean = ls[0] * inv_cnt;
    const float var  = ls2[0] * inv_cnt - mean * mean;
    const float sc   = gamma[c] * rsqrtf(var + BN_EPS);
    scale[c] = sc;
    shift[c] = beta[c] - mean * sc;
  }
}

// ---------------------------------------------------------------------------
// Kernel 3: ball query. Thread = one query point (b,n). Candidates tiled
// through LDS. Keep first K candidate indices (ascending m), pad with first.
// Self-distance is 0 <= R^2 so cnt >= 1 always.
// ---------------------------------------------------------------------------
__global__ void __launch_bounds__(256) ball_query_k(const float* __restrict__ p,
                                                    int* __restrict__ idx) {
  const int gid = blockIdx.x * 256 + threadIdx.x;  // 0..B*N-1, whole block same b
  const int b   = gid / N_;
  const int n   = gid % N_;
  const float* __restrict__ pb = p + (size_t)b * N_ * 3;
  const float qx = pb[n * 3 + 0];
  const float qy = pb[n * 3 + 1];
  const float qz = pb[n * 3 + 2];

  __shared__ float sx[256], sy[256], sz[256];
  int* __restrict__ out = idx + (size_t)gid * K_;
  int cnt = 0;
  int first = n;

  for (int t = 0; t < N_; t += 256) {
    const int m = t + threadIdx.x;
    sx[threadIdx.x] = pb[m * 3 + 0];
    sy[threadIdx.x] = pb[m * 3 + 1];
    sz[threadIdx.x] = pb[m * 3 + 2];
    __syncthreads();
    if (cnt < K_) {
      for (int j = 0; j < 256; ++j) {
        const float dx = sx[j] - qx, dy = sy[j] - qy, dz = sz[j] - qz;
        const float d2 = dx * dx + dy * dy + dz * dz;
        if (d2 <= RADIUS2 && cnt < K_) {
          if (cnt == 0) first = t + j;
          out[cnt] = t + j;
          ++cnt;
        }
      }
    }
    __syncthreads();
  }
  for (int k = cnt; k < K_; ++k) out[k] = first;
}

// ---------------------------------------------------------------------------
// Kernel 4 (memory-bound main pass):
//   out[b,c,n] = max_k( dlp[b,c,n,k] + relu(h[b,c,idx[b,n,k]]*scale[c]+shift[c]) )
// Block = (b, 32-wide n tile). idx tile staged once in LDS, reused for all C.
// dlp streamed with non-temporal 128-bit loads (single use; keep h L2-hot).
// ---------------------------------------------------------------------------
__global__ void __launch_bounds__(256) fuse_gather_max(const float* __restrict__ dlp,
                                                       const float* __restrict__ h,
                                                       const int* __restrict__ idx,
                                                       const float* __restrict__ scale,
                                                       const float* __restrict__ shift,
                                                       float* __restrict__ out) {
  const int ntiles = N_ / 32;
  const int b  = blockIdx.x / ntiles;
  const int n0 = (blockIdx.x % ntiles) * 32;
  const int tid = threadIdx.x;

  __shared__ int sidx[32 * K_];  // 4 KB
  const int* __restrict__ itile = idx + ((size_t)b * N_ + n0) * K_;
  for (int i = tid; i < 32 * K_; i += 256) sidx[i] = itile[i];
  __syncthreads();

  const int nl = tid & 31;     // n within tile  -> coalesced dim
  const int cg = tid >> 5;     // 0..7 channel group
  const int n  = n0 + nl;
  const int* __restrict__ ip = &sidx[nl * K_];

#pragma unroll 2
  for (int ci = 0; ci < C_ / 8; ++ci) {
    const int c = ci * 8 + cg;
    const float s  = scale[c];
    const float sh = shift[c];
    const float* __restrict__ hrow = h + ((size_t)b * C_ + c) * N_;
    const float* __restrict__ dptr = dlp + (((size_t)b * C_ + c) * N_ + n) * (size_t)K_;

    float m = -__builtin_inff();
#pragma unroll
    for (int k4 = 0; k4 < K_; k4 += 4) {
      const v4f d = __builtin_nontemporal_load((const v4f*)(dptr + k4));
      const float g0 = fmaxf(0.f, fmaf(hrow[ip[k4 + 0]], s, sh));
      const float g1 = fmaxf(0.f, fmaf(hrow[ip[k4 + 1]], s, sh));
      const float g2 = fmaxf(0.f, fmaf(hrow[ip[k4 + 2]], s, sh));
      const float g3 = fmaxf(0.f, fmaf(hrow[ip[k4 + 3]], s, sh));
      m = fmaxf(m, d.x + g0);
      m = fmaxf(m, d.y + g1);
      m = fmaxf(m, d.z + g2);
      m = fmaxf(m, d.w + g3);
    }
    out[((size_t)b * C_ + c) * N_ + n] = m;
  }
}

// ---------------------------------------------------------------------------
// Launch. Inputs (setup_inputs order): p, f, dlp, W1, b1, gamma, beta.
// Workspace: h (8 MB) | idx (2 MB) | scale/shift (1 KB).
// ---------------------------------------------------------------------------
extern "C" void kernel_launch(void* const* d_in, const int* in_sizes, int n_in,
                              void* d_out, int out_size, void* d_ws, size_t ws_size,
                              hipStream_t stream) {
  const float* p     = (const float*)d_in[0];
  const float* f     = (const float*)d_in[1];
  const float* dlp   = (const float*)d_in[2];
  const float* W1    = (const float*)d_in[3];
  const float* b1    = (const float*)d_in[4];
  const float* gamma = (const float*)d_in[5];
  const float* beta  = (const float*)d_in[6];
  float* out = (float*)d_out;

  const size_t H_BYTES   = (size_t)B_ * C_ * N_ * sizeof(float);   // 8 MB
  const size_t IDX_BYTES = (size_t)B_ * N_ * K_ * sizeof(int);     // 2 MB
  float* h     = (float*)d_ws;
  int*   idx   = (int*)((char*)d_ws + H_BYTES);
  float* scale = (float*)((char*)d_ws + H_BYTES + IDX_BYTES);
  float* shift = scale + C_;

  // 1) GEMM via f32 WMMA: one wave per 16x16 tile.
  gemm_wmma<<<dim3(N_ / 16, C_ / 16, B_), 32, 0, stream>>>(W1, f, b1, h);
  // 2) BN statistics -> fused scale/shift.
  bn_stats<<<C_, 256, 0, stream>>>(h, gamma, beta, scale, shift);
  // 3) Ball query (independent of 1&2; stream order keeps it pipelined).
  ball_query_k<<<(B_ * N_) / 256, 256, 0, stream>>>(p, idx);
  // 4) Fused gather + affine + relu + add + max over K.
  fuse_gather_max<<<B_ * (N_ / 32), 256, 0, stream>>>(dlp, h, idx, scale, shift, out);
}